// GCN_27393301414235
// MI455X (gfx1250) — compile-verified
//
#include <hip/hip_runtime.h>

#define NNODES 100000
#define NEDGES 1600000
#define BN_EPS 1e-5

typedef float v2f __attribute__((ext_vector_type(2)));
typedef float v8f __attribute__((ext_vector_type(8)));

__global__ void zero_f32_kernel(float* __restrict__ p, int n) {
    int i = blockIdx.x * blockDim.x + threadIdx.x;
    int stride = gridDim.x * blockDim.x;
    for (; i < n; i += stride) p[i] = 0.0f;
}

__global__ void degree_kernel(const int* __restrict__ src, const int* __restrict__ dst,
                              float* __restrict__ degO, float* __restrict__ degI, int e_total) {
    int i = blockIdx.x * blockDim.x + threadIdx.x;
    int stride = gridDim.x * blockDim.x;
    for (; i < e_total; i += stride) {
        unsafeAtomicAdd(&degO[src[i]], 1.0f);
        unsafeAtomicAdd(&degI[dst[i]], 1.0f);
    }
}

__global__ void norm_kernel(float* __restrict__ degO, float* __restrict__ degI, int n) {
    int i = blockIdx.x * blockDim.x + threadIdx.x;
    if (i < n) {
        degO[i] = rsqrtf(fmaxf(degO[i], 1.0f));
        degI[i] = rsqrtf(fmaxf(degI[i], 1.0f));
    }
}

// One wave32 per edge: lane covers 4 consecutive features (float4 gather, 4 f32 atomics).
// Fuses the n_out[src] scaling of the reference.
__global__ void __launch_bounds__(256) scatter_kernel(
    const float* __restrict__ x, const int* __restrict__ src, const int* __restrict__ dst,
    const float* __restrict__ nOut, float* __restrict__ agg, int e_total) {
    int e = blockIdx.x * (blockDim.x >> 5) + (threadIdx.x >> 5);
    if (e >= e_total) return;
    int lane = threadIdx.x & 31;
    int s = src[e], d = dst[e];
    float ns = nOut[s];
    const float4* row = (const float4*)(x + (size_t)s * 128);
    float4 v = row[lane];
    float* outp = agg + (size_t)d * 128 + lane * 4;
    unsafeAtomicAdd(outp + 0, v.x * ns);
    unsafeAtomicAdd(outp + 1, v.y * ns);
    unsafeAtomicAdd(outp + 2, v.z * ns);
    unsafeAtomicAdd(outp + 3, v.w * ns);
}

// out[row0..row0+15][0..NCOLS) = (agg * n_in[row]) @ W + bias, exact fp32 via
// V_WMMA_F32_16X16X4_F32. One wave per 16x16 output tile; NCOLS/16 waves per block.
// If STATS: accumulate per-column sum / sum-of-squares (fp64 atomics) for BN.
template <int NCOLS, bool STATS>
__global__ void __launch_bounds__(NCOLS * 2) gemm_kernel(
    const float* __restrict__ A, const float* __restrict__ nIn,
    const float* __restrict__ W, const float* __restrict__ bias,
    float* __restrict__ out, double* __restrict__ stats) {
    __shared__ float As[16 * 130];   // stride 130 -> conflict-free fragment reads
    const int row0 = blockIdx.x * 16;
    const int tid = threadIdx.x;
    #pragma unroll
    for (int i = tid; i < 16 * 128; i += NCOLS * 2) {
        int m = i >> 7, k = i & 127;
        As[m * 130 + k] = A[(size_t)(row0 + m) * 128 + k] * nIn[row0 + m];
    }
    __syncthreads();

    const int lane = tid & 31;
    const int wave = tid >> 5;
    const int half = lane >> 4;       // which K-pair / row-half this lane owns
    const int mcol = lane & 15;
    const int col  = wave * 16 + mcol;

    v8f acc = {0.f, 0.f, 0.f, 0.f, 0.f, 0.f, 0.f, 0.f};
    const float* as = &As[mcol * 130 + half * 2];
    const float* wp = &W[(size_t)(half * 2) * NCOLS + col];
    #pragma unroll
    for (int k = 0; k < 128; k += 4) {
        v2f a, b;
        a.x = as[k];
        a.y = as[k + 1];
        b.x = wp[(size_t)k * NCOLS];
        b.y = wp[(size_t)(k + 1) * NCOLS];
        acc = __builtin_amdgcn_wmma_f32_16x16x4_f32(false, a, false, b,
                                                    (short)0, acc, false, false);
    }

    float bv = bias[col];
    #pragma unroll
    for (int r = 0; r < 8; ++r) acc[r] += bv;

    #pragma unroll
    for (int r = 0; r < 8; ++r) {
        int row = row0 + half * 8 + r;
        out[(size_t)row * NCOLS + col] = acc[r];
    }

    if (STATS) {
        float s = 0.f, q = 0.f;
        #pragma unroll
        for (int r = 0; r < 8; ++r) { s += acc[r]; q += acc[r] * acc[r]; }
        s += __shfl_xor(s, 16, 32);
        q += __shfl_xor(q, 16, 32);
        if (half == 0) {
            unsafeAtomicAdd(&stats[col], (double)s);
            unsafeAtomicAdd(&stats[128 + col], (double)q);
        }
    }
}

__global__ void bn_finalize_kernel(const double* __restrict__ stats,
                                   const float* __restrict__ g, const float* __restrict__ be,
                                   float* __restrict__ colScale, float* __restrict__ colShift,
                                   int n_rows) {
    int j = threadIdx.x;
    double inv = 1.0 / (double)n_rows;
    double mean = stats[j] * inv;
    double var = stats[128 + j] * inv - mean * mean;
    double sc = (double)g[j] * rsqrt(var + (double)BN_EPS);
    colScale[j] = (float)sc;
    colShift[j] = (float)((double)be[j] - mean * sc);
}

__global__ void bn_relu_kernel(float4* __restrict__ h, const float4* __restrict__ colScale4,
                               const float4* __restrict__ colShift4, int n4) {
    int i = blockIdx.x * blockDim.x + threadIdx.x;
    int stride = gridDim.x * blockDim.x;
    for (; i < n4; i += stride) {
        int j = i & 31;                 // 32 float4 per 128-wide row
        float4 v = h[i], sc = colScale4[j], sh = colShift4[j];
        v.x = fmaxf(v.x * sc.x + sh.x, 0.f);
        v.y = fmaxf(v.y * sc.y + sh.y, 0.f);
        v.z = fmaxf(v.z * sc.z + sh.z, 0.f);
        v.w = fmaxf(v.w * sc.w + sh.w, 0.f);
        h[i] = v;
    }
}

extern "C" void kernel_launch(void* const* d_in, const int* in_sizes, int n_in,
                              void* d_out, int out_size, void* d_ws, size_t ws_size,
                              hipStream_t stream) {
    const float* features = (const float*)d_in[0];
    const int*   src      = (const int*)d_in[1];
    const int*   dst      = (const int*)d_in[2];
    const float* W1 = (const float*)d_in[3];
    const float* b1 = (const float*)d_in[4];
    const float* g1 = (const float*)d_in[5];
    const float* be1 = (const float*)d_in[6];
    const float* W2 = (const float*)d_in[7];
    const float* b2 = (const float*)d_in[8];
    const float* g2 = (const float*)d_in[9];
    const float* be2 = (const float*)d_in[10];
    const float* W3 = (const float*)d_in[11];
    const float* b3 = (const float*)d_in[12];
    float* out = (float*)d_out;

    const int N = NNODES, E = NEDGES;
    const int NF = N * 128;

    char* ws = (char*)d_ws;
    float*  agg      = (float*)(ws);                    // 51.2 MB
    float*  h        = (float*)(ws + 51200000);         // 51.2 MB
    float*  nOut     = (float*)(ws + 102400000);        // 0.4 MB
    float*  nIn      = (float*)(ws + 102800000);        // 0.4 MB
    double* stats    = (double*)(ws + 103200000);       // 2 KB (8B aligned)
    float*  colScale = (float*)(ws + 103202048);
    float*  colShift = (float*)(ws + 103202560);

    // ---- degree norms ----
    zero_f32_kernel<<<512, 256, 0, stream>>>(nOut, N);
    zero_f32_kernel<<<512, 256, 0, stream>>>(nIn, N);
    degree_kernel<<<2048, 256, 0, stream>>>(src, dst, nOut, nIn, E);
    norm_kernel<<<(N + 255) / 256, 256, 0, stream>>>(nOut, nIn, N);

    dim3 sgrid(E / 8);   // 8 waves (edges) per 256-thread block

    // ---- layer 1 ----
    zero_f32_kernel<<<4096, 256, 0, stream>>>(agg, NF);
    scatter_kernel<<<sgrid, 256, 0, stream>>>(features, src, dst, nOut, agg, E);
    zero_f32_kernel<<<2, 256, 0, stream>>>((float*)stats, 512);
    gemm_kernel<128, true><<<N / 16, 256, 0, stream>>>(agg, nIn, W1, b1, h, stats);
    bn_finalize_kernel<<<1, 128, 0, stream>>>(stats, g1, be1, colScale, colShift, N);
    bn_relu_kernel<<<4096, 256, 0, stream>>>((float4*)h, (const float4*)colScale,
                                             (const float4*)colShift, N * 32);

    // ---- layer 2 ----
    zero_f32_kernel<<<4096, 256, 0, stream>>>(agg, NF);
    scatter_kernel<<<sgrid, 256, 0, stream>>>(h, src, dst, nOut, agg, E);
    zero_f32_kernel<<<2, 256, 0, stream>>>((float*)stats, 512);
    gemm_kernel<128, true><<<N / 16, 256, 0, stream>>>(agg, nIn, W2, b2, h, stats);
    bn_finalize_kernel<<<1, 128, 0, stream>>>(stats, g2, be2, colScale, colShift, N);
    bn_relu_kernel<<<4096, 256, 0, stream>>>((float4*)h, (const float4*)colScale,
                                             (const float4*)colShift, N * 32);

    // ---- layer 3 (64 outputs, straight to d_out) ----
    zero_f32_kernel<<<4096, 256, 0, stream>>>(agg, NF);
    scatter_kernel<<<sgrid, 256, 0, stream>>>(h, src, dst, nOut, agg, E);
    gemm_kernel<64, false><<<N / 16, 128, 0, stream>>>(agg, nIn, W3, b3, out, nullptr);
}